// cha_second_conv_20040317403397
// MI455X (gfx1250) — compile-verified
//
#include <hip/hip_runtime.h>

typedef float v2f __attribute__((ext_vector_type(2)));
typedef float v8f __attribute__((ext_vector_type(8)));

#define C_IN 64
#define FEAT 224        // 16 groups * 14 features
#define OUTC 64
#define HW   16384      // 128*128
#define PIX_PER_BLK 64
#define FSTRIDE 226     // padded per-pixel feature stride (dwords), bank-conflict-free
#define K4STEPS 56      // FEAT / 4

// Pre-pack W (64x224 row-major, fp32) into WMMA A-fragment lane order,
// k-major so per-k-step fragments are contiguous (immediate-offset friendly):
// Wf[k4][m][lane] = { W[m*16 + lane%16][4*k4 + 2*(lane/16)],
//                     W[m*16 + lane%16][4*k4 + 2*(lane/16) + 1] }
__global__ void pack_w_kernel(const float* __restrict__ Wg, float* __restrict__ Wf) {
    int e = blockIdx.x * blockDim.x + threadIdx.x;   // 0 .. 56*4*32-1
    if (e >= K4STEPS * 4 * 32) return;
    int lane = e & 31;
    int m    = (e >> 5) & 3;
    int k4   = e >> 7;
    int o    = m * 16 + (lane & 15);
    int col  = 4 * k4 + 2 * (lane >> 4);
    Wf[2 * e]     = Wg[o * FEAT + col];
    Wf[2 * e + 1] = Wg[o * FEAT + col + 1];
}

__global__ __launch_bounds__(128) void cha_conv_kernel(
        const float* __restrict__ inp, const float* __restrict__ Wf,
        const float* __restrict__ bias, float* __restrict__ out) {
    __shared__ float feat[PIX_PER_BLK * FSTRIDE];   // 57,856 B

    const int t   = threadIdx.x;
    const int b   = blockIdx.x >> 8;          // 256 pixel-blocks per image
    const int blk = blockIdx.x & 255;
    const int pixbase = blk * PIX_PER_BLK;    // flat pixel offset within HW plane

    // ---- Phase 1: build 224-wide feature vectors for 64 pixels into LDS ----
    {
        const int p  = t & 63;                // local pixel
        const int gh = t >> 6;                // 0/1: which half of the 16 groups
        const float* ip0 = inp + (size_t)b * C_IN * HW + pixbase + p;
        #pragma unroll
        for (int i = 0; i < 8; ++i) {
            const int g = gh * 8 + i;
            const float* ip = ip0 + (size_t)(4 * g) * HW;
            float x0 = ip[0 * HW], x1 = ip[1 * HW], x2 = ip[2 * HW], x3 = ip[3 * HW];
            float* fp = &feat[p * FSTRIDE + g * 14];
            fp[0]  = x0;      fp[1]  = x1;      fp[2]  = x2;      fp[3]  = x3;
            fp[4]  = x0 * x0; fp[5]  = x1 * x1; fp[6]  = x2 * x2; fp[7]  = x3 * x3;
            fp[8]  = x0 * x1; fp[9]  = x1 * x2; fp[10] = x2 * x3;   // j=1 pairs
            fp[11] = x0 * x2; fp[12] = x1 * x3;                     // j=2 pairs
            fp[13] = x0 * x3;                                       // j=3 pair
        }
    }
    __syncthreads();

    // ---- Phase 2: 2 M-tiles x 2 N-tiles per wave, V_WMMA_F32_16X16X4_F32 ----
    const int lane = t & 31;
    const int wv   = t >> 5;                  // wave id 0..3
    const int mh   = wv >> 1;                 // M-half: output rows {32mh..32mh+31}
    const int nh   = wv & 1;                  // N-half: pixels {32nh..32nh+31}
    const int pr   = lane & 15;
    const int hi   = lane >> 4;               // half-wave selector (K +2 / M +8)

    v8f acc[2][2] = {};                       // [mi][ni] 16x16 f32 tiles
    // Wave-fixed A base: both per-step frags at +k4*1024B and +k4*1024B+256B
    const v2f* wfb = (const v2f*)Wf + (2 * mh) * 32 + lane;
    const float* fl0 = &feat[(nh * 32 + pr) * FSTRIDE + 2 * hi];
    const float* fl1 = fl0 + 16 * FSTRIDE;

    for (int k4 = 0; k4 < K4STEPS; ++k4) {
        v2f bf0 = *(const v2f*)(fl0 + 4 * k4);   // B frag, N-tile 0
        v2f bf1 = *(const v2f*)(fl1 + 4 * k4);   // B frag, N-tile 1
        v2f a0  = wfb[k4 * 128];                 // A frag, M-tile 2mh
        v2f a1  = wfb[k4 * 128 + 32];            // A frag, M-tile 2mh+1
        acc[0][0] = __builtin_amdgcn_wmma_f32_16x16x4_f32(
            false, a0, false, bf0, (short)0, acc[0][0], false, false);
        acc[0][1] = __builtin_amdgcn_wmma_f32_16x16x4_f32(
            false, a0, false, bf1, (short)0, acc[0][1], false, false);
        acc[1][0] = __builtin_amdgcn_wmma_f32_16x16x4_f32(
            false, a1, false, bf0, (short)0, acc[1][0], false, false);
        acc[1][1] = __builtin_amdgcn_wmma_f32_16x16x4_f32(
            false, a1, false, bf1, (short)0, acc[1][1], false, false);
    }

    // ---- Epilogue: bias add + coalesced stores ----
    #pragma unroll
    for (int mi = 0; mi < 2; ++mi) {
        const int oc0 = (2 * mh + mi) * 16 + 8 * hi;   // D: VGPR r -> M = r + 8*hi
        #pragma unroll
        for (int ni = 0; ni < 2; ++ni) {
            const size_t pbase = (size_t)b * OUTC * HW + pixbase
                               + nh * 32 + ni * 16 + pr;
            #pragma unroll
            for (int r = 0; r < 8; ++r) {
                const int oc = oc0 + r;
                out[pbase + (size_t)oc * HW] = acc[mi][ni][r] + bias[oc];
            }
        }
    }
}

extern "C" void kernel_launch(void* const* d_in, const int* in_sizes, int n_in,
                              void* d_out, int out_size, void* d_ws, size_t ws_size,
                              hipStream_t stream) {
    const float* inp  = (const float*)d_in[0];   // (16,64,128,128) f32
    const float* Wg   = (const float*)d_in[1];   // (64,224) f32
    const float* bias = (const float*)d_in[2];   // (64,) f32
    float* out = (float*)d_out;                  // (16,64,128,128) f32
    float* Wf  = (float*)d_ws;                   // 56*4*32*2 floats = 57,344 B

    const int nent = K4STEPS * 4 * 32;
    pack_w_kernel<<<(nent + 255) / 256, 256, 0, stream>>>(Wg, Wf);

    cha_conv_kernel<<<dim3(16 * 256), 128, 0, stream>>>(inp, Wf, bias, out);
}